// DualMemorySystem_6193342841009
// MI455X (gfx1250) — compile-verified
//
#include <hip/hip_runtime.h>
#include <hip/hip_bf16.h>

typedef float v2f __attribute__((ext_vector_type(2)));
typedef float v8f __attribute__((ext_vector_type(8)));

namespace {
constexpr int kB = 2, kC = 16, kH = 128, kW = 128;
constexpr int kHW = kH * kW;          // 16384
constexpr int kL = kB * kHW;          // 32768 rows
constexpr int kBCHW = kB * kC * kHW;  // 524288
}

__device__ __forceinline__ v8f wmma_f32(v2f a, v2f b, v8f c) {
  // D = A(16x4 f32) * B(4x16 f32) + C(16x16 f32)
  return __builtin_amdgcn_wmma_f32_16x16x4_f32(false, a, false, b, (short)0, c,
                                               false, false);
}

__device__ __forceinline__ float fold_div(int Y, int X, int pad) {
  int cy = min(Y, pad) + min(kH - 1 - Y, pad) + 1;
  int cx = min(X, pad) + min(kW - 1 - X, pad) + 1;
  return (float)(cy * cx) + 1e-8f;
}

// ---------------------------------------------------------------------------
// Phase 1: att[r, m] = softmax_m( unfold(x)[r,:] . mem[m,:] * temp/sqrt(D) )
// One wave computes a 16-row x 16-slot sim tile via V_WMMA_F32_16X16X4_F32.
// K dimension is reordered as (i, j, 16-channel blocks) so every operand
// address in the hot loop is base + compile-time immediate offset.
// ---------------------------------------------------------------------------
template <int P, int MS, int NT>
__global__ __launch_bounds__(NT * 32) void att_kernel(
    const float* __restrict__ x, const float* __restrict__ mem,
    const float* __restrict__ temp, float* __restrict__ att) {
  constexpr int pad = P / 2;
  constexpr int PATW = 16 + 2 * pad;
  constexpr int PP = P * P;
  constexpr int D = kC * PP;
  constexpr int CSTR = P * PATW;  // channel stride in patch

  __shared__ float patch[kC * P * PATW];
  __shared__ float red[NT][16];

  const int row0 = blockIdx.x * 16;
  const int b = row0 / kHW;
  const int l0 = row0 % kHW;
  const int y = l0 / kW;
  const int x0 = l0 % kW;  // tile = 16 consecutive pixels in one image row

  // Stage zero-padded input patch: [C][P rows][PATW cols]
  for (int e = threadIdx.x; e < kC * P * PATW; e += NT * 32) {
    int c = e / (P * PATW);
    int r = e % (P * PATW);
    int i = r / PATW;
    int xx = r % PATW;
    int yy = y + i - pad;
    int gx = x0 + xx - pad;
    float v = 0.0f;
    if (yy >= 0 && yy < kH && gx >= 0 && gx < kW)
      v = x[((b * kC + c) * kH + yy) * kW + gx];
    patch[e] = v;
  }
  __syncthreads();

  const int lane = threadIdx.x & 31;
  const int wv = threadIdx.x >> 5;
  const int half = lane >> 4;  // K slot pair: 0 -> {0,1}, 1 -> {2,3}
  const int ln = lane & 15;    // A: pixel row m,  B: slot column n
  const int k0 = half * 2;
  const int n0 = wv * 16;

  // B row pointer; out-of-range tg slots clamped (their sim columns are
  // masked to -inf before softmax, so garbage values are harmless).
  const int slot = min(n0 + ln, MS - 1);
  const float* __restrict__ mrow = mem + (size_t)slot * D;

  v8f acc = {0.f, 0.f, 0.f, 0.f, 0.f, 0.f, 0.f, 0.f};

  int aoff = ln + k0 * CSTR;  // patch index: (c=k0)*CSTR + i*PATW + j + ln
  int boff = k0 * PP;         // mem-row index: (c=k0)*PP + i*P + j
  for (int i = 0; i < P; ++i) {
    for (int j = 0; j < P; ++j) {
#pragma unroll
      for (int step = 0; step < 4; ++step) {  // channels step*4+k0, +1
        v2f a, bb;
        a.x = patch[aoff + step * 4 * CSTR];
        a.y = patch[aoff + step * 4 * CSTR + CSTR];
        bb.x = mrow[boff + step * 4 * PP];
        bb.y = mrow[boff + step * 4 * PP + PP];
        acc = wmma_f32(a, bb, acc);
      }
      aoff += 1;
      boff += 1;
    }
    aoff += PATW - P;
  }

  const float scale = temp[0] * rsqrtf((float)D);
#pragma unroll
  for (int r = 0; r < 8; ++r) acc[r] *= scale;
  if (MS < NT * 16) {  // mask padded slot columns before softmax (tg)
    if (n0 + ln >= MS) {
#pragma unroll
      for (int r = 0; r < 8; ++r) acc[r] = -__builtin_inff();
    }
  }

  // Row-wise max over 16-lane halves (C-layout: row m = r + 8*half, col = ln)
  float rmax[8], rsum[8];
#pragma unroll
  for (int r = 0; r < 8; ++r) {
    float v = acc[r];
    v = fmaxf(v, __shfl_xor(v, 1, 32));
    v = fmaxf(v, __shfl_xor(v, 2, 32));
    v = fmaxf(v, __shfl_xor(v, 4, 32));
    v = fmaxf(v, __shfl_xor(v, 8, 32));
    rmax[r] = v;
  }
  if (NT > 1) {
    if (ln == 0) {
#pragma unroll
      for (int r = 0; r < 8; ++r) red[wv][half * 8 + r] = rmax[r];
    }
    __syncthreads();
#pragma unroll
    for (int r = 0; r < 8; ++r) {
      float v = -__builtin_inff();
      for (int w2 = 0; w2 < NT; ++w2) v = fmaxf(v, red[w2][half * 8 + r]);
      rmax[r] = v;
    }
    __syncthreads();
  }
#pragma unroll
  for (int r = 0; r < 8; ++r) {
    float e = __expf(acc[r] - rmax[r]);
    acc[r] = e;
    float v = e;
    v += __shfl_xor(v, 1, 32);
    v += __shfl_xor(v, 2, 32);
    v += __shfl_xor(v, 4, 32);
    v += __shfl_xor(v, 8, 32);
    rsum[r] = v;
  }
  if (NT > 1) {
    if (ln == 0) {
#pragma unroll
      for (int r = 0; r < 8; ++r) red[wv][half * 8 + r] = rsum[r];
    }
    __syncthreads();
#pragma unroll
    for (int r = 0; r < 8; ++r) {
      float v = 0.f;
      for (int w2 = 0; w2 < NT; ++w2) v += red[w2][half * 8 + r];
      rsum[r] = v;
    }
  }
#pragma unroll
  for (int r = 0; r < 8; ++r) {
    int m = half * 8 + r;
    int col = n0 + ln;
    if (col < MS) att[(size_t)(row0 + m) * MS + col] = acc[r] / rsum[r];
  }
}

// ---------------------------------------------------------------------------
// Phase 2: fused read-GEMM + fold:
//   out[b,c,Y,X] = sum_{i,j} sum_m att[b, (Y+pad-i)W+(X+pad-j), m]
//                                 * mem[m, c*P*P + i*P + j]
// One wave per 16-pixel tile: p*p GEMMs of [16 x M] x [M x 16 ch], att halo
// staged in LDS, incremental affine addressing.  Overlap division deferred.
// ---------------------------------------------------------------------------
template <int P, int MS>
__global__ __launch_bounds__(32) void fold_kernel(const float* __restrict__ att,
                                                  const float* __restrict__ mem,
                                                  float* __restrict__ feat) {
  constexpr int pad = P / 2;
  constexpr int PATW = 16 + 2 * pad;
  constexpr int PP = P * P;
  constexpr int D = kC * PP;

  __shared__ float satt[P * PATW * MS];

  const int row0 = blockIdx.x * 16;
  const int b = row0 / kHW;
  const int l0 = row0 % kHW;
  const int Y = l0 / kW;
  const int x0 = l0 % kW;

  // Stage att halo: rows y' in [Y-pad, Y+pad], cols x' in [x0-pad, x0+15+pad]
  for (int e = threadIdx.x; e < P * PATW * MS; e += 32) {
    int pos = e / MS;
    int m = e % MS;
    int dy = pos / PATW;
    int dx = pos % PATW;
    int yy = Y - pad + dy;
    int gx = x0 - pad + dx;
    float v = 0.f;
    if (yy >= 0 && yy < kH && gx >= 0 && gx < kW)
      v = att[(size_t)(b * kHW + yy * kW + gx) * MS + m];
    satt[e] = v;
  }
  __syncthreads();

  const int lane = threadIdx.x & 31;
  const int half = lane >> 4;
  const int ln = lane & 15;  // A: pixel m, B: channel c
  const int k0 = half * 2;

  v8f acc = {0.f, 0.f, 0.f, 0.f, 0.f, 0.f, 0.f, 0.f};

  int bbase = ln * PP + k0 * D;  // mem index: + k (linear over i,j), + kk*D
  for (int i = 0; i < P; ++i) {
    int abase = ((P - 1 - i) * PATW + ln + (P - 1)) * MS + k0;
    for (int j = 0; j < P; ++j) {
#pragma unroll
      for (int kk = 0; kk < MS; kk += 4) {
        v2f a, bb;
        a.x = satt[abase + kk];
        a.y = satt[abase + kk + 1];
        bb.x = mem[bbase + kk * D];
        bb.y = mem[bbase + kk * D + D];
        acc = wmma_f32(a, bb, acc);
      }
      abase -= MS;
      bbase += 1;
    }
  }

#pragma unroll
  for (int r = 0; r < 8; ++r) {
    int m = half * 8 + r;  // pixel within tile
    feat[((b * kC + ln) * kH + Y) * kW + (x0 + m)] = acc[r];
  }
}

// ---------------------------------------------------------------------------
// Phase 3a: pooled[branch,b,c] = mean_{Y,X} sum_s feat_s / div_s
// ---------------------------------------------------------------------------
__global__ __launch_bounds__(256) void pool_kernel(
    const float* __restrict__ feat, float* __restrict__ pooled) {
  const int gid = blockIdx.x;  // branch*32 + b*16 + c
  const int branch = gid >> 5;
  const int bc = gid & 31;
  const float* f0 = feat + (size_t)(branch * 3 + 0) * kBCHW + (size_t)bc * kHW;
  const float* f1 = feat + (size_t)(branch * 3 + 1) * kBCHW + (size_t)bc * kHW;
  const float* f2 = feat + (size_t)(branch * 3 + 2) * kBCHW + (size_t)bc * kHW;
  float s = 0.f;
  for (int p = threadIdx.x; p < kHW; p += 256) {
    int Y = p >> 7, X = p & 127;
    s += f0[p] / fold_div(Y, X, 1) + f1[p] / fold_div(Y, X, 2) +
         f2[p] / fold_div(Y, X, 3);
  }
  __shared__ float sh[256];
  sh[threadIdx.x] = s;
  __syncthreads();
  for (int o = 128; o > 0; o >>= 1) {
    if (threadIdx.x < o) sh[threadIdx.x] += sh[threadIdx.x + o];
    __syncthreads();
  }
  if (threadIdx.x == 0) pooled[gid] = sh[0] * (1.0f / kHW);
}

// ---------------------------------------------------------------------------
// Phase 3b: tiny 2-layer MLP + softmax over scales -> wt[branch][b][s][c]
// ---------------------------------------------------------------------------
__global__ __launch_bounds__(128) void mlp_kernel(
    const float* __restrict__ pooled, const float* __restrict__ w1,
    const float* __restrict__ b1, const float* __restrict__ w2,
    const float* __restrict__ b2, int branch, float* __restrict__ wt) {
  __shared__ float sp[kB][kC];
  __shared__ float shd[kB][4];
  __shared__ float sl[kB][48];
  int t = threadIdx.x;
  if (t < kB * kC) sp[t / kC][t % kC] = pooled[branch * kB * kC + t];
  __syncthreads();
  if (t < kB * 4) {
    int b = t >> 2, h = t & 3;
    float a = b1[h];
    for (int c = 0; c < kC; ++c) a += w1[h * kC + c] * sp[b][c];
    shd[b][h] = fmaxf(a, 0.f);
  }
  __syncthreads();
  if (t < kB * 48) {
    int b = t / 48, o = t % 48;
    float a = b2[o];
    for (int h = 0; h < 4; ++h) a += w2[o * 4 + h] * shd[b][h];
    sl[b][o] = a;
  }
  __syncthreads();
  if (t < kB * kC) {
    int b = t >> 4, c = t & 15;
    float l0 = sl[b][c], l1 = sl[b][16 + c], l2 = sl[b][32 + c];
    float mx = fmaxf(l0, fmaxf(l1, l2));
    float e0 = __expf(l0 - mx), e1 = __expf(l1 - mx), e2 = __expf(l2 - mx);
    float inv = 1.f / (e0 + e1 + e2);
    int base = branch * 96 + b * 48;
    wt[base + c] = e0 * inv;
    wt[base + 16 + c] = e1 * inv;
    wt[base + 32 + c] = e2 * inv;
  }
}

// ---------------------------------------------------------------------------
// Phase 3c: out = sum_s wt[branch,b,s,c] * feat_s / div_s
// ---------------------------------------------------------------------------
__global__ __launch_bounds__(256) void final_kernel(
    const float* __restrict__ feat, const float* __restrict__ wt,
    float* __restrict__ out) {
  size_t idx = (size_t)blockIdx.x * 256 + threadIdx.x;  // < 2*kBCHW
  int branch = (int)(idx / kBCHW);
  int r = (int)(idx % kBCHW);
  int b = r / (kC * kHW);
  int rc = r % (kC * kHW);
  int c = rc / kHW;
  int p = rc % kHW;
  int Y = p >> 7, X = p & 127;
  const float* fb = feat + (size_t)branch * 3 * kBCHW + r;
  int wbase = branch * 96 + b * 48;
  float v = wt[wbase + c] * fb[0] / fold_div(Y, X, 1) +
            wt[wbase + 16 + c] * fb[(size_t)kBCHW] / fold_div(Y, X, 2) +
            wt[wbase + 32 + c] * fb[2 * (size_t)kBCHW] / fold_div(Y, X, 3);
  out[idx] = v;
}

// ---------------------------------------------------------------------------
extern "C" void kernel_launch(void* const* d_in, const int* in_sizes, int n_in,
                              void* d_out, int out_size, void* d_ws,
                              size_t ws_size, hipStream_t stream) {
  (void)in_sizes; (void)n_in; (void)out_size; (void)ws_size;
  const float* bg = (const float*)d_in[0];
  const float* tg = (const float*)d_in[1];
  const float *bg_mem[3], *tg_mem[3], *bg_temp[3], *tg_temp[3];
  for (int s = 0; s < 3; ++s) {
    bg_mem[s] = (const float*)d_in[2 + 4 * s + 0];
    tg_mem[s] = (const float*)d_in[2 + 4 * s + 1];
    bg_temp[s] = (const float*)d_in[2 + 4 * s + 2];
    tg_temp[s] = (const float*)d_in[2 + 4 * s + 3];
  }
  const float* bg_fc1_w = (const float*)d_in[14];
  const float* bg_fc1_b = (const float*)d_in[15];
  const float* bg_fc2_w = (const float*)d_in[16];
  const float* bg_fc2_b = (const float*)d_in[17];
  const float* tg_fc1_w = (const float*)d_in[18];
  const float* tg_fc1_b = (const float*)d_in[19];
  const float* tg_fc2_w = (const float*)d_in[20];
  const float* tg_fc2_b = (const float*)d_in[21];

  float* ws = (float*)d_ws;
  float* att_bg = ws;                       // 3 * kL * 64
  float* att_tg = att_bg + 3ull * kL * 64;  // 3 * kL * 8
  float* feat = att_tg + 3ull * kL * 8;     // 6 * kBCHW  [branch*3+s]
  float* pooled = feat + 6ull * kBCHW;      // 64
  float* wt = pooled + 64;                  // 192

  const int tiles = kL / 16;  // 2048

  // scale 0: P=3
  att_kernel<3, 64, 4><<<tiles, 128, 0, stream>>>(bg, bg_mem[0], bg_temp[0],
                                                  att_bg + 0ull * kL * 64);
  att_kernel<3, 8, 1><<<tiles, 32, 0, stream>>>(tg, tg_mem[0], tg_temp[0],
                                                att_tg + 0ull * kL * 8);
  fold_kernel<3, 64><<<tiles, 32, 0, stream>>>(att_bg + 0ull * kL * 64,
                                               bg_mem[0], feat + 0ull * kBCHW);
  fold_kernel<3, 8><<<tiles, 32, 0, stream>>>(att_tg + 0ull * kL * 8, tg_mem[0],
                                              feat + 3ull * kBCHW);
  // scale 1: P=5
  att_kernel<5, 64, 4><<<tiles, 128, 0, stream>>>(bg, bg_mem[1], bg_temp[1],
                                                  att_bg + 1ull * kL * 64);
  att_kernel<5, 8, 1><<<tiles, 32, 0, stream>>>(tg, tg_mem[1], tg_temp[1],
                                                att_tg + 1ull * kL * 8);
  fold_kernel<5, 64><<<tiles, 32, 0, stream>>>(att_bg + 1ull * kL * 64,
                                               bg_mem[1], feat + 1ull * kBCHW);
  fold_kernel<5, 8><<<tiles, 32, 0, stream>>>(att_tg + 1ull * kL * 8, tg_mem[1],
                                              feat + 4ull * kBCHW);
  // scale 2: P=7
  att_kernel<7, 64, 4><<<tiles, 128, 0, stream>>>(bg, bg_mem[2], bg_temp[2],
                                                  att_bg + 2ull * kL * 64);
  att_kernel<7, 8, 1><<<tiles, 32, 0, stream>>>(tg, tg_mem[2], tg_temp[2],
                                                att_tg + 2ull * kL * 8);
  fold_kernel<7, 64><<<tiles, 32, 0, stream>>>(att_bg + 2ull * kL * 64,
                                               bg_mem[2], feat + 2ull * kBCHW);
  fold_kernel<7, 8><<<tiles, 32, 0, stream>>>(att_tg + 2ull * kL * 8, tg_mem[2],
                                              feat + 5ull * kBCHW);

  pool_kernel<<<64, 256, 0, stream>>>(feat, pooled);
  mlp_kernel<<<1, 128, 0, stream>>>(pooled, bg_fc1_w, bg_fc1_b, bg_fc2_w,
                                    bg_fc2_b, 0, wt);
  mlp_kernel<<<1, 128, 0, stream>>>(pooled, tg_fc1_w, tg_fc1_b, tg_fc2_w,
                                    tg_fc2_b, 1, wt);
  final_kernel<<<(2 * kBCHW) / 256, 256, 0, stream>>>(feat, wt, (float*)d_out);
}